// StereoDenoiser_28449863368800
// MI455X (gfx1250) — compile-verified
//
#include <hip/hip_runtime.h>
#include <cstdint>

// ---------------------------------------------------------------------------
// StereoDenoiser for MI455X (gfx1250).
// Memory-bound (~0.82 GB HBM traffic, ~35 us at 23.3 TB/s peak). Strategy:
// stage halo'd row segments into LDS with CDNA5 async global->LDS copies
// (B128 fast path, clamped B32 at image borders), then do the disparity-
// dependent bilinear gathers out of LDS.
// ---------------------------------------------------------------------------

#define AS_GLOBAL __attribute__((address_space(1)))
#define AS_LDS    __attribute__((address_space(3)))

typedef int v4i __attribute__((vector_size(4 * sizeof(int))));

namespace {
constexpr int   kW      = 1920;
constexpr int   kH      = 1024;
constexpr int   kC      = 3;
constexpr int   kB      = 8;
constexpr int   kTile   = 640;           // 1920 = 3 * 640, no partial tiles
constexpr int   kHalo   = 128;           // disparity bound in pixels
constexpr int   kSeg    = kTile + 132;   // staged floats per segment (mult. of 4)
constexpr int   kNSeg   = 6;             // img_r c0..2 then img_l c0..2
constexpr int   kGroups = kSeg / 4;      // 193 float4-groups per segment
constexpr float kMaxDisp = 128.0f;
}

#if defined(__HIP_DEVICE_COMPILE__) && __has_builtin(__builtin_amdgcn_global_load_async_to_lds_b32)
#define USE_ASYNC_LDS 1
#else
#define USE_ASYNC_LDS 0
#endif

#if USE_ASYNC_LDS && __has_builtin(__builtin_amdgcn_global_load_async_to_lds_b128)
#define USE_ASYNC_B128 1
#else
#define USE_ASYNC_B128 0
#endif

__global__ __launch_bounds__(kTile)
void stereo_denoise_kernel(const float* __restrict__ img_l,
                           const float* __restrict__ img_r,
                           const float* __restrict__ depth,
                           float* __restrict__ out_l,
                           float* __restrict__ out_r)
{
    __shared__ float lds[kNSeg * kSeg];   // 18528 bytes

    const int tid  = (int)threadIdx.x;
    const int tile = (int)blockIdx.x;
    const int h    = (int)blockIdx.y;
    const int b    = (int)blockIdx.z;
    const int x0   = tile * kTile;
    const int r0   = x0 - kHalo;          // base column of img_r segments
    const int l0   = x0;                  // base column of img_l segments

    // ---- stage halo'd row segments (16B async groups; clamped at borders) ----
    for (int g = tid; g < kNSeg * kGroups; g += kTile) {
        const int  seg  = g / kGroups;
        const int  j    = g - seg * kGroups;
        const int  i    = 4 * j;
        const bool isR  = seg < 3;
        const int  c    = isR ? seg : seg - 3;
        const int  base = isR ? r0 : l0;
        const int  col  = base + i;
        const float* row = (isR ? img_r : img_l)
                         + ((size_t)((b * kC + c) * kH + h)) * kW;
        float* dst = &lds[seg * kSeg + i];

        if (col >= 0 && col <= kW - 4) {
            // fully in-bounds, 16B-aligned group
#if USE_ASYNC_B128
            __builtin_amdgcn_global_load_async_to_lds_b128(
                (AS_GLOBAL v4i*)(row + col), (AS_LDS v4i*)dst, 0, 0);
#elif USE_ASYNC_LDS
#pragma unroll
            for (int e = 0; e < 4; ++e)
                __builtin_amdgcn_global_load_async_to_lds_b32(
                    (AS_GLOBAL int*)(row + col + e), (AS_LDS int*)(dst + e), 0, 0);
#else
#pragma unroll
            for (int e = 0; e < 4; ++e) dst[e] = row[col + e];
#endif
        } else {
            // border group: clamp each element (replicated edge pixels)
#pragma unroll
            for (int e = 0; e < 4; ++e) {
                int ce = col + e;
                ce = ce < 0 ? 0 : (ce > kW - 1 ? kW - 1 : ce);
#if USE_ASYNC_LDS
                __builtin_amdgcn_global_load_async_to_lds_b32(
                    (AS_GLOBAL int*)(row + ce), (AS_LDS int*)(dst + e), 0, 0);
#else
                dst[e] = row[ce];
#endif
            }
        }
    }

#if USE_ASYNC_LDS
#if __has_builtin(__builtin_amdgcn_s_wait_asynccnt)
    __builtin_amdgcn_s_wait_asynccnt(0);
#else
    asm volatile("s_wait_asynccnt 0" ::: "memory");
#endif
#endif
    __syncthreads();

    // ---- per-pixel disparity + bilinear taps (match reference math) ----
    const int   xx   = x0 + tid;
    const float d    = depth[((size_t)(b * kH + h)) * kW + xx];
    const float disp = fminf(fmaxf(d * kMaxDisp, 0.0f), kMaxDisp);

    // warp of right image: x = xx - disp
    float xr = fminf(fmaxf((float)xx - disp, 0.0f), (float)(kW - 1));
    float fr = floorf(xr);
    float wr = xr - fr;
    int   i0r = (int)fr;
    int   i1r = i0r + 1 < kW - 1 ? i0r + 1 : kW - 1;
    int   a0r = i0r - r0;
    int   a1r = i1r - r0;

    // warp of left image: x = xx + disp
    float xl = fminf(fmaxf((float)xx + disp, 0.0f), (float)(kW - 1));
    float fl = floorf(xl);
    float wl = xl - fl;
    int   i0l = (int)fl;
    int   i1l = i0l + 1 < kW - 1 ? i0l + 1 : kW - 1;
    int   a0l = i0l - l0;
    int   a1l = i1l - l0;

    const size_t pix0 = ((size_t)((b * kC) * kH + h)) * kW + xx;   // c = 0 offset

#pragma unroll
    for (int c = 0; c < kC; ++c) {
        const float* ldsR = &lds[c * kSeg];          // img_r segment
        const float* ldsL = &lds[(c + 3) * kSeg];    // img_l segment

        const float warped_r = ldsR[a0r] * (1.0f - wr) + ldsR[a1r] * wr;
        const float warped_l = ldsL[a0l] * (1.0f - wl) + ldsL[a1l] * wl;
        const float il = ldsL[tid];                  // img_l[xx] (staged)
        const float ir = ldsR[tid + kHalo];          // img_r[xx] (staged)

        const size_t o = pix0 + (size_t)c * ((size_t)kH * kW);
        out_l[o] = (il + warped_r) * 0.5f;
        out_r[o] = (ir + warped_l) * 0.5f;
    }
}

extern "C" void kernel_launch(void* const* d_in, const int* in_sizes, int n_in,
                              void* d_out, int out_size, void* d_ws, size_t ws_size,
                              hipStream_t stream)
{
    (void)in_sizes; (void)n_in; (void)out_size; (void)d_ws; (void)ws_size;

    const float* img_l = (const float*)d_in[0];
    const float* img_r = (const float*)d_in[1];
    const float* depth = (const float*)d_in[2];

    float* out_l = (float*)d_out;
    float* out_r = out_l + (size_t)kB * kC * kH * kW;

    dim3 grid(kW / kTile, kH, kB);   // (3, 1024, 8)
    stereo_denoise_kernel<<<grid, kTile, 0, stream>>>(img_l, img_r, depth,
                                                      out_l, out_r);
}